// RCModel_87643102642705
// MI455X (gfx1250) — compile-verified
//
#include <hip/hip_runtime.h>
#include <hip/hip_bf16.h>

// ---------------------------------------------------------------------------
// RC thermal model RK4: collapsed to x_{n+1} = Phi x_n + a*Tg[n] + b*Tg[n+1] + g
// Chunked (S=16) so output generation becomes per-batch GEMMs done with
// V_WMMA_F32_16X16X4_F32. Serial part reduced to a 511-step chunk scan.
// ---------------------------------------------------------------------------

#define NS      10      // states
#define NR      8       // rooms
#define TSTEPS  8192
#define NBATCH  512
#define SCH     16      // chunk length
#define NCH     512     // number of chunks (last one ragged: 15 valid steps)
#define MROWS   160     // SCH * NS output rows per chunk
#define KA      16      // padded K for CfA * Y   (10 -> 16)
#define KB      20      // padded K for CfB * TgZ (17+1 -> 20)
#define KREC    28      // 10 + 17 + 1
#define DT_H    60.0f
#define H6      10.0f   // h/6

// workspace layout (floats)
#define OFS_CFA  0u
#define SZ_CFA   (NBATCH * MROWS * KA)                 // 1,310,720
#define OFS_CFB  (OFS_CFA + SZ_CFA)
#define SZ_CFB   (NBATCH * MROWS * KB)                 // 1,638,400
#define OFS_CREC (OFS_CFB + SZ_CFB)
#define SZ_CREC  (NBATCH * NS * KREC)                  //   143,360
#define OFS_Y    (OFS_CREC + SZ_CREC)
#define SZ_Y     (NBATCH * KA * NCH)                   // 4,194,304
#define OFS_TGZ  (OFS_Y + SZ_Y)
#define SZ_TGZ   (KB * NCH)                            //    10,240

typedef float v2f __attribute__((ext_vector_type(2)));
typedef float v8f __attribute__((ext_vector_type(8)));

__device__ __forceinline__ float sigm(float x) { return 1.0f / (1.0f + __expf(-x)); }

// ---------------------------------------------------------------------------
// Kernel 1: per-batch coefficient construction (one 256-thread block / batch)
// ---------------------------------------------------------------------------
__global__ __launch_bounds__(256) void prep_kernel(
    const float* __restrict__ A, const float* __restrict__ Bm,
    const float* __restrict__ loads, const float* __restrict__ areas,
    float* __restrict__ ws)
{
    const int b = blockIdx.x;
    const int t = threadIdx.x;

    __shared__ float H[100], H2[100], H3[100], H4[100];
    __shared__ float Phi[100], P0[100], Pm[100], Ecur[100], Enx[100];
    __shared__ float av[NS], bv[NS], gv[NS], cv[NS], b0[NS], Q[NR], Gacc[NS];
    __shared__ float alpha[SCH][NS], beta[SCH][NS];

    if (t < 100) H[t] = DT_H * A[b * 100 + t];
    if (t < NR) {
        float l0 = loads[b * 2 * NR + t];
        float l1 = loads[b * 2 * NR + NR + t];
        Q[t] = (sigm(l1) - sigm(l0)) * 100.0f * areas[t];
    }
    if (t < NS) b0[t] = Bm[b * NS * (1 + NR) + t * (1 + NR)];
    __syncthreads();

    if (t < 100) {
        int i = t / 10, j = t % 10; float s = 0.f;
        #pragma unroll
        for (int k = 0; k < 10; ++k) s += H[i * 10 + k] * H[k * 10 + j];
        H2[t] = s;
    }
    __syncthreads();
    if (t < 100) {
        int i = t / 10, j = t % 10; float s3 = 0.f, s4 = 0.f;
        #pragma unroll
        for (int k = 0; k < 10; ++k) {
            s3 += H2[i * 10 + k] * H[k * 10 + j];
            s4 += H2[i * 10 + k] * H2[k * 10 + j];
        }
        H3[t] = s3; H4[t] = s4;
    }
    if (t < NS) {  // c = B[:,:,1:] @ Q
        float s = 0.f;
        #pragma unroll
        for (int r = 0; r < NR; ++r) s += Bm[b * NS * (1 + NR) + t * (1 + NR) + 1 + r] * Q[r];
        cv[t] = s;
    }
    __syncthreads();
    if (t < 100) {
        int i = t / 10, j = t % 10; float id = (i == j) ? 1.0f : 0.0f;
        Phi[t] = id + H[t] + 0.5f * H2[t] + H3[t] / 6.0f + H4[t] / 24.0f;
        P0[t]  = H6 * (id + H[t] + 0.5f * H2[t] + 0.25f * H3[t]);
        Pm[t]  = H6 * (4.0f * id + 2.0f * H[t] + 0.5f * H2[t]);
    }
    __syncthreads();
    if (t < NS) {
        float s1 = 0.f, s2 = 0.f, s3 = 0.f;
        #pragma unroll
        for (int m = 0; m < 10; ++m) {
            s1 += (P0[t * 10 + m] + 0.5f * Pm[t * 10 + m]) * b0[m];
            s2 += Pm[t * 10 + m] * b0[m];
            s3 += (P0[t * 10 + m] + Pm[t * 10 + m]) * cv[m];
        }
        av[t] = s1;
        bv[t] = H6 * b0[t] + 0.5f * s2;   // P1 = (h/6) I
        gv[t] = s3 + H6 * cv[t];
        Gacc[t] = 0.0f;
    }
    if (t < 100) Ecur[t] = (t / 10 == t % 10) ? 1.0f : 0.0f;  // E_0 = I
    __syncthreads();

    float* CfA  = ws + OFS_CFA  + (size_t)b * MROWS * KA;
    float* CfB  = ws + OFS_CFB  + (size_t)b * MROWS * KB;
    float* Crec = ws + OFS_CREC + (size_t)b * NS * KREC;

    for (int d = 0; d <= SCH; ++d) {
        if (d >= 1) {
            if (t < 160) {                       // CfA rows for j=d  <- E_d
                int s = t / 16, m = t % 16;
                CfA[((d - 1) * NS + s) * KA + m] = (m < 10) ? Ecur[s * 10 + m] : 0.0f;
            }
            if (t < NS) CfB[((d - 1) * NS + t) * KB + 17] = Gacc[t];   // Gsum_j
            if (d == SCH && t < NS) {
                #pragma unroll
                for (int m = 0; m < 10; ++m) Crec[t * KREC + m] = Ecur[t * 10 + m]; // Phi^16
                Crec[t * KREC + 27] = Gacc[t];
            }
        }
        __syncthreads();
        if (d < SCH) {
            if (t < NS) {
                float sa = 0.f, sb = 0.f, sg = 0.f;
                #pragma unroll
                for (int m = 0; m < 10; ++m) {
                    float e = Ecur[t * 10 + m];
                    sa += e * av[m]; sb += e * bv[m]; sg += e * gv[m];
                }
                alpha[d][t] = sa; beta[d][t] = sb; Gacc[t] += sg;
            }
            if (t < 100) {
                int i = t / 10, j = t % 10; float s = 0.f;
                #pragma unroll
                for (int k = 0; k < 10; ++k) s += Phi[i * 10 + k] * Ecur[k * 10 + j];
                Enx[t] = s;
            }
            __syncthreads();
            if (t < 100) Ecur[t] = Enx[t];
            __syncthreads();
        }
    }
    __syncthreads();

    // CfB columns 0..16 (Tout-grid coefficients), 18,19 pad
    for (int idx = t; idx < MROWS * KB; idx += blockDim.x) {
        int r = idx / KB, m = idx % KB;
        if (m == 17) continue;                      // written above
        int j = r / NS + 1, s = r % NS;
        float val = 0.f;
        if (m <= 16) {
            if (m <= j - 1) val += alpha[j - 1 - m][s];
            if (m >= 1 && m <= j) val += beta[j - m][s];
        }
        CfB[r * KB + m] = val;
    }
    // Crec columns 10..26 (kappa for j=16)
    for (int idx = t; idx < NS * 17; idx += blockDim.x) {
        int s = idx / 17, m = idx % 17;
        float val = 0.f;
        if (m <= 15) val += alpha[15 - m][s];
        if (m >= 1)  val += beta[16 - m][s];
        Crec[s * KREC + 10 + m] = val;
    }
}

// ---------------------------------------------------------------------------
// Kernel 2: shared Tout chunk matrix TgZ (KB x NCH)
// ---------------------------------------------------------------------------
__global__ __launch_bounds__(256) void tgz_kernel(const float* __restrict__ Tg,
                                                  float* __restrict__ ws)
{
    int idx = blockIdx.x * blockDim.x + threadIdx.x;
    if (idx >= KB * NCH) return;
    int m = idx / NCH, k = idx % NCH;
    float v;
    if (m <= 16) {
        int gi = k * SCH + m;
        if (gi > TSTEPS - 1) gi = TSTEPS - 1;       // clamped value only feeds the
        v = Tg[gi];                                 // discarded j=16 row of chunk 511
    } else if (m == 17) v = 1.0f;
    else v = 0.0f;
    ws[OFS_TGZ + m * NCH + k] = v;
}

// ---------------------------------------------------------------------------
// Kernel 3: chunk-level scan, one wave per batch:
//   y_{k+1} = Crec * [y_k ; Tg17_k ; 1]
// ---------------------------------------------------------------------------
__global__ __launch_bounds__(32) void scan_kernel(const float* __restrict__ Tg,
                                                  const float* __restrict__ x0,
                                                  float* __restrict__ ws,
                                                  float* __restrict__ out)
{
    const int b = blockIdx.x;
    const int lane = threadIdx.x;
    const float* Crec = ws + OFS_CREC + (size_t)b * NS * KREC;

    float crec[KREC];
    #pragma unroll
    for (int m = 0; m < KREC; ++m) crec[m] = (lane < NS) ? Crec[lane * KREC + m] : 0.0f;

    __shared__ float z[KREC];
    if (lane < NS) {
        float v = x0[b * NS + lane];
        z[lane] = v;
        out[b * NS + lane] = v;                     // output row 0 = x0
    }
    __syncthreads();

    float* Yb = ws + OFS_Y + (size_t)b * KA * NCH;  // row-major [state][chunk]
    for (int k = 0; k < NCH; ++k) {
        if (lane < KA) Yb[lane * NCH + k] = (lane < NS) ? z[lane] : 0.0f;
        if (k < NCH - 1) {
            if (lane < 17) z[NS + lane] = Tg[k * SCH + lane];
            if (lane == 17) z[27] = 1.0f;
            __syncthreads();
            float acc = 0.f;
            if (lane < NS) {
                #pragma unroll
                for (int m = 0; m < KREC; ++m) acc += crec[m] * z[m];
            }
            __syncthreads();
            if (lane < NS) z[lane] = acc;
            __syncthreads();
        }
    }
}

// ---------------------------------------------------------------------------
// Kernel 4: output GEMM, one wave per 16x16 tile:
//   D(160x512) = CfA(160x16) * Y(16x512) + CfB(160x20) * TgZ(20x512)
// 9 x V_WMMA_F32_16X16X4_F32 per tile, then scatter-store to (8192,512,10).
// A/B operand layout per ISA: lane = (M|N) + 16*floor(K/2); reg0<->K=2*hi,
// reg1<->K=2*hi+1.
// ---------------------------------------------------------------------------
__global__ __launch_bounds__(32) void gemm_kernel(const float* __restrict__ ws,
                                                  float* __restrict__ out)
{
    const int blk = blockIdx.x;
    const int b   = blk / 320;
    const int tt  = blk % 320;
    const int mt  = tt / 32;          // 10 M-tiles
    const int nt  = tt % 32;          // 32 N-tiles
    const int lane = threadIdx.x;
    const int lm = lane & 15;
    const int lh = lane >> 4;

    const float* CfA = ws + OFS_CFA + (size_t)b * MROWS * KA;
    const float* CfB = ws + OFS_CFB + (size_t)b * MROWS * KB;
    const float* Yb  = ws + OFS_Y   + (size_t)b * KA * NCH;
    const float* Tz  = ws + OFS_TGZ;

    const int rowA = mt * 16 + lm;
    const int colN = nt * 16 + lm;

    v8f acc = {};
    #pragma unroll
    for (int q = 0; q < 4; ++q) {                  // CfA * Y   (K = 16)
        int k0 = 4 * q + 2 * lh;
        v2f a, bb;
        a.x  = CfA[rowA * KA + k0];
        a.y  = CfA[rowA * KA + k0 + 1];
        bb.x = Yb[(k0) * NCH + colN];
        bb.y = Yb[(k0 + 1) * NCH + colN];
        acc = __builtin_amdgcn_wmma_f32_16x16x4_f32(false, a, false, bb,
                                                    (short)0, acc, false, false);
    }
    #pragma unroll
    for (int q = 0; q < 5; ++q) {                  // CfB * TgZ (K = 20)
        int k0 = 4 * q + 2 * lh;
        v2f a, bb;
        a.x  = CfB[rowA * KB + k0];
        a.y  = CfB[rowA * KB + k0 + 1];
        bb.x = Tz[(k0) * NCH + colN];
        bb.y = Tz[(k0 + 1) * NCH + colN];
        acc = __builtin_amdgcn_wmma_f32_16x16x4_f32(false, a, false, bb,
                                                    (short)0, acc, false, false);
    }

    // D layout: VGPR p, lanes 0-15 -> M=p, lanes 16-31 -> M=8+p; N = lane%16
    #pragma unroll
    for (int p = 0; p < 8; ++p) {
        int M = p + 8 * lh;
        int r = mt * 16 + M;
        int j = r / NS + 1;
        int s = r - (j - 1) * NS;
        int grow = colN * SCH + j;                 // global time index
        if (grow < TSTEPS)
            out[(size_t)grow * (NBATCH * NS) + b * NS + s] = acc[p];
    }
}

// ---------------------------------------------------------------------------
extern "C" void kernel_launch(void* const* d_in, const int* in_sizes, int n_in,
                              void* d_out, int out_size, void* d_ws, size_t ws_size,
                              hipStream_t stream)
{
    (void)in_sizes; (void)n_in; (void)out_size; (void)ws_size;
    const float* Tg    = (const float*)d_in[1];
    const float* A     = (const float*)d_in[2];
    const float* Bm    = (const float*)d_in[3];
    const float* loads = (const float*)d_in[4];
    const float* areas = (const float*)d_in[5];
    const float* x0    = (const float*)d_in[6];
    float* ws  = (float*)d_ws;
    float* out = (float*)d_out;

    prep_kernel<<<NBATCH, 256, 0, stream>>>(A, Bm, loads, areas, ws);
    tgz_kernel<<<(KB * NCH + 255) / 256, 256, 0, stream>>>(Tg, ws);
    scan_kernel<<<NBATCH, 32, 0, stream>>>(Tg, x0, ws, out);
    gemm_kernel<<<NBATCH * 320, 32, 0, stream>>>(ws, out);
}